// GRAFFLayer_86388972191749
// MI455X (gfx1250) — compile-verified
//
#include <hip/hip_runtime.h>

typedef __attribute__((ext_vector_type(2))) float v2f;
typedef __attribute__((ext_vector_type(8))) float v8f;

#define DIM 128
#define STEPSZ 0.1f

// ---------------------------------------------------------------------------
// Zero the ax accumulator (aliased with d_out) and the degree buffer.
// ---------------------------------------------------------------------------
__global__ void graff_zero(float* __restrict__ ax, float* __restrict__ deg, int n) {
  long long total  = (long long)n * DIM;
  long long stride = (long long)gridDim.x * blockDim.x;
  for (long long i = (long long)blockIdx.x * blockDim.x + threadIdx.x; i < total; i += stride)
    ax[i] = 0.0f;
  for (long long i = (long long)blockIdx.x * blockDim.x + threadIdx.x; i < n; i += stride)
    deg[i] = 0.0f;
}

// ---------------------------------------------------------------------------
// In-degree via atomics (one thread per edge).
// ---------------------------------------------------------------------------
__global__ void graff_deg(const int* __restrict__ dst, float* __restrict__ deg, int E) {
  int e = blockIdx.x * blockDim.x + threadIdx.x;
  if (e < E) atomicAdd(&deg[dst[e]], 1.0f);
}

// dinv = 1/sqrt(deg + 1)  (self-loop adds 1)
__global__ void graff_dinv(const float* __restrict__ deg, float* __restrict__ dinv, int n) {
  int i = blockIdx.x * blockDim.x + threadIdx.x;
  if (i < n) dinv[i] = rsqrtf(deg[i] + 1.0f);
}

// wstarT[n][k] = 0.5 * (W[k][n] + W[n][k])   (TRANSPOSED symmetrized weight)
// Transposed layout makes the WMMA B-fragment's two K-values contiguous, so
// the GEMM inner loop loads B with one global_load_b64 instead of two b32.
__global__ void graff_wstar(const float* __restrict__ W, float* __restrict__ wstarT) {
  int i = blockIdx.x * blockDim.x + threadIdx.x;  // 16384 threads
  int nrow = i >> 7, k = i & (DIM - 1);           // wstarT row = n, col = k
  wstarT[i] = 0.5f * (W[k * DIM + nrow] + W[nrow * DIM + k]);
}

// ---------------------------------------------------------------------------
// Edge scatter: one wave per edge. Lane l handles features l, l+32, l+64, l+96.
// Gathers of x[src] and atomic adds to ax[dst] are wave-coalesced; both
// buffers (51 MB each) are L2-resident on MI455X (192 MB L2).
// ---------------------------------------------------------------------------
__global__ void graff_scatter(const float* __restrict__ x,
                              const int* __restrict__ src,
                              const int* __restrict__ dst,
                              const float* __restrict__ dinv,
                              float* __restrict__ ax, int E) {
  int gid  = blockIdx.x * blockDim.x + threadIdx.x;
  int e    = gid >> 5;          // wave-uniform -> scalarized loads
  int lane = gid & 31;
  if (e >= E) return;
  int s = src[e];
  int d = dst[e];
  float ew = dinv[s] * dinv[d];
  const float* xr = x + (long long)s * DIM;
  float*       ar = ax + (long long)d * DIM;
#pragma unroll
  for (int j = 0; j < 4; ++j) {
    int c = lane + 32 * j;
    atomicAdd(&ar[c], xr[c] * ew);
  }
}

// ---------------------------------------------------------------------------
// Fused epilogue GEMM using V_WMMA_F32_16X16X4_F32:
//   out = x + (STEP * (ax_scatter + x*dinv^2)) @ w_star
// Block = 256 threads (8 waves). Block computes a 16-row stripe; wave w owns
// the 16x16 tile at columns [16w, 16w+16). A-stripe staged in LDS via float4
// (stride 132 floats keeps 16B alignment AND spreads banks); B streamed as
// b64 loads from the transposed, L2-resident wstarT. ax aliases out: rows are
// fully read into LDS (barrier) before being overwritten, and each 16-row
// stripe is owned by exactly one block -> safe in-place.
// ---------------------------------------------------------------------------
__global__ void graff_gemm(const float* __restrict__ x,
                           const float* ax,            // aliases out
                           const float* __restrict__ dinv,
                           const float* __restrict__ wstarT,
                           float* out, int n) {
  __shared__ float sA[16][DIM + 4];  // stride 132 floats

  const int block_row = blockIdx.x * 16;
  const int tid  = threadIdx.x;
  const int wave = tid >> 5;
  const int lane = tid & 31;

  // Stage s = STEP * (ax + x*dinv^2) for 16 rows: 512 float4s, 2 per thread.
  for (int i = tid; i < 16 * (DIM / 4); i += 256) {
    int r  = i >> 5;                 // 32 float4 per row
    int c4 = (i & 31) * 4;
    int row = block_row + r;
    if (row >= n) row = n - 1;       // clamp (N=100000 is 16-aligned anyway)
    long long g = (long long)row * DIM + c4;
    float di  = dinv[row];
    float di2 = di * di;
    float4 av = *(const float4*)(ax + g);
    float4 xv = *(const float4*)(x + g);
    sA[r][c4 + 0] = STEPSZ * (av.x + xv.x * di2);
    sA[r][c4 + 1] = STEPSZ * (av.y + xv.y * di2);
    sA[r][c4 + 2] = STEPSZ * (av.z + xv.z * di2);
    sA[r][c4 + 3] = STEPSZ * (av.w + xv.w * di2);
  }
  __syncthreads();

  const int ncol = wave * 16;            // this wave's output columns
  const int arow = lane & 15;            // A: M = lane (both halves)
  const int koff = (lane >> 4) * 2;      // A/B: K = {0,1} or {2,3} per half-wave
  const int bn   = ncol + (lane & 15);   // B: N striped across lanes
  const float* brow = wstarT + (long long)bn * DIM;  // lane's B column, contiguous in K

  v8f acc = {};
#pragma unroll
  for (int k = 0; k < DIM; k += 4) {
    v2f a, b;
    a[0] = sA[arow][k + koff];
    a[1] = sA[arow][k + koff + 1];
    b = *(const v2f*)(brow + k + koff);  // 8B-aligned: one global_load_b64
    acc = __builtin_amdgcn_wmma_f32_16x16x4_f32(
        /*neg_a=*/false, a, /*neg_b=*/false, b,
        /*c_mod=*/(short)0, acc, /*reuse_a=*/false, /*reuse_b=*/false);
  }

  // D layout: VGPR r -> M = r (lanes 0-15) or 8+r (lanes 16-31), N = lane&15.
  const int mbase = (lane >> 4) * 8;
  const int col   = ncol + (lane & 15);
#pragma unroll
  for (int r = 0; r < 8; ++r) {
    int row = block_row + mbase + r;
    if (row < n)
      out[(long long)row * DIM + col] = x[(long long)row * DIM + col] + acc[r];
  }
}

// ---------------------------------------------------------------------------
extern "C" void kernel_launch(void* const* d_in, const int* in_sizes, int n_in,
                              void* d_out, int out_size, void* d_ws, size_t ws_size,
                              hipStream_t stream) {
  const float* x  = (const float*)d_in[0];
  const float* W  = (const float*)d_in[1];
  const int*   ei = (const int*)d_in[2];

  float* out = (float*)d_out;
  const int n = in_sizes[0] / DIM;     // 100000
  const int E = in_sizes[2] / 2;       // 1600000
  const int* src = ei;
  const int* dst = ei + E;

  // Workspace layout: deg[n] | dinv[n] | wstarT[128*128]  (< 1 MB)
  float* deg    = (float*)d_ws;
  float* dinv   = deg + n;
  float* wstarT = dinv + n;

  // 1. zero ax (== d_out) and deg
  graff_zero<<<4096, 256, 0, stream>>>(out, deg, n);
  // 2. in-degree
  graff_deg<<<(E + 255) / 256, 256, 0, stream>>>(dst, deg, E);
  // 3. dinv
  graff_dinv<<<(n + 255) / 256, 256, 0, stream>>>(deg, dinv, n);
  // 4. wstarT = transpose of 0.5(W + W^T)  (symmetric, but stored n-major)
  graff_wstar<<<(DIM * DIM) / 256, 256, 0, stream>>>(W, wstarT);
  // 5. edge scatter into d_out (one wave per edge, 8 edges per block)
  graff_scatter<<<(E + 7) / 8, 256, 0, stream>>>(x, src, dst, dinv, out, E);
  // 6. fused self-loop + GEMM + residual, in-place on d_out
  graff_gemm<<<(n + 15) / 16, 256, 0, stream>>>(x, out, dinv, wstarT, out, n);
}